// GCN_67877663146444
// MI455X (gfx1250) — compile-verified
//
#include <hip/hip_runtime.h>

typedef __attribute__((ext_vector_type(2))) float v2f;
typedef __attribute__((ext_vector_type(8))) float v8f;

#define GEPS 1e-5f

// ---------------- elementwise / utility kernels ----------------

__global__ void fill0(float* __restrict__ p, long n) {
  long i = (long)blockIdx.x * blockDim.x + threadIdx.x;
  if (i < n) p[i] = 0.f;
}

__global__ void deg_accum(const int* __restrict__ dst, float* __restrict__ deg, long E) {
  long e = (long)blockIdx.x * blockDim.x + threadIdx.x;
  if (e < E) atomicAdd(&deg[dst[e]], 1.0f);
}

__global__ void dinv_k(float* __restrict__ d, long N) {
  long i = (long)blockIdx.x * blockDim.x + threadIdx.x;
  if (i < N) d[i] = rsqrtf(d[i] + 1.0f);
}

__global__ void count_nodes(const int* __restrict__ batch, float* __restrict__ gcnt, long N) {
  long i = (long)blockIdx.x * blockDim.x + threadIdx.x;
  if (i < N) atomicAdd(&gcnt[batch[i]], 1.0f);
}

// ---------------- GCN conv pieces ----------------

// h[N,128] = x[N,7] @ W[7,128]; one block per node, 128 threads.
__global__ void gemm_in7(const float* __restrict__ x, const float* __restrict__ W,
                         float* __restrict__ h, long N) {
  __shared__ float xr[8];
  long i = blockIdx.x;
  int j = threadIdx.x;
  if (j < 7) xr[j] = x[i * 7 + j];
  __syncthreads();
  float acc = 0.f;
#pragma unroll
  for (int k = 0; k < 7; ++k) acc += xr[k] * W[k * 128 + j];
  h[i * 128 + j] = acc;
}

// One wave (32 lanes) per edge; lane handles 4 contiguous columns.
__global__ void scatter_edges(const int* __restrict__ src, const int* __restrict__ dst,
                              const float* __restrict__ dinv, const float* __restrict__ h,
                              float* __restrict__ agg, long E) {
  long t = (long)blockIdx.x * blockDim.x + threadIdx.x;
  long e = t >> 5;
  int lane = (int)(t & 31);
  if (e >= E) return;
  int s = src[e], d = dst[e];
  float coef = dinv[s] * dinv[d];
  float4 v = *((const float4*)(h + (long)s * 128) + lane);
  float* ap = agg + (long)d * 128 + lane * 4;
  atomicAdd(ap + 0, v.x * coef);
  atomicAdd(ap + 1, v.y * coef);
  atomicAdd(ap + 2, v.z * coef);
  atomicAdd(ap + 3, v.w * coef);
}

// agg += h * dinv^2 + b
__global__ void selfloop_bias(const float* __restrict__ h, const float* __restrict__ dinv,
                              const float* __restrict__ b, float* __restrict__ agg, long N) {
  long t = (long)blockIdx.x * blockDim.x + threadIdx.x;
  if (t >= N * 128) return;
  long i = t >> 7;
  int j = (int)(t & 127);
  float di = dinv[i];
  agg[t] += h[t] * di * di + b[j];
}

// ---------------- GraphNorm pieces ----------------

// segment-sum rows (optionally squared) into acc[G,128]; one wave per node.
__global__ void seg_accum(const float* __restrict__ xin, const int* __restrict__ batch,
                          float* __restrict__ acc, long N, int square) {
  long t = (long)blockIdx.x * blockDim.x + threadIdx.x;
  long i = t >> 5;
  int lane = (int)(t & 31);
  if (i >= N) return;
  int g = batch[i];
  float4 v = *((const float4*)(xin + i * 128) + lane);
  if (square) { v.x *= v.x; v.y *= v.y; v.z *= v.z; v.w *= v.w; }
  float* ap = acc + (long)g * 128 + lane * 4;
  atomicAdd(ap + 0, v.x);
  atomicAdd(ap + 1, v.y);
  atomicAdd(ap + 2, v.z);
  atomicAdd(ap + 3, v.w);
}

__global__ void mean_div(const float* __restrict__ gsum, const float* __restrict__ gcnt,
                         float* __restrict__ gmean, long G) {
  long t = (long)blockIdx.x * blockDim.x + threadIdx.x;
  if (t >= G * 128) return;
  gmean[t] = gsum[t] / fmaxf(gcnt[t >> 7], 1.0f);
}

__global__ void center_k(const float* __restrict__ xin, const float* __restrict__ gmean,
                         const float* __restrict__ alpha, const int* __restrict__ batch,
                         float* __restrict__ out, long N) {
  long t = (long)blockIdx.x * blockDim.x + threadIdx.x;
  if (t >= N * 128) return;
  long i = t >> 7;
  int j = (int)(t & 127);
  int g = batch[i];
  out[t] = xin[t] - alpha[j] * gmean[(long)g * 128 + j];
}

// y = relu(w * c / sqrt(var+eps) + b)
__global__ void gnorm_final(const float* __restrict__ c, const float* __restrict__ vsum,
                            const float* __restrict__ gcnt, const float* __restrict__ w,
                            const float* __restrict__ b, const int* __restrict__ batch,
                            float* __restrict__ out, long N) {
  long t = (long)blockIdx.x * blockDim.x + threadIdx.x;
  if (t >= N * 128) return;
  long i = t >> 7;
  int j = (int)(t & 127);
  int g = batch[i];
  float cnt = fmaxf(gcnt[g], 1.0f);
  float var = vsum[(long)g * 128 + j] / cnt;
  float y = w[j] * c[t] * rsqrtf(var + GEPS) + b[j];
  out[t] = fmaxf(y, 0.f);
}

__global__ void pool_div(float* __restrict__ pooled, const float* __restrict__ gcnt, long G) {
  long t = (long)blockIdx.x * blockDim.x + threadIdx.x;
  if (t >= G * 128) return;
  pooled[t] /= fmaxf(gcnt[t >> 7], 1.0f);
}

// ---------------- fp32 WMMA GEMM: C[M,128] = A[M,128] @ B[128,128] ----------------
// Block = 256 threads = 8 waves; block owns a 16-row stripe; wave w owns cols [16w,16w+16).
// A tile (16x128) is staged once per block in LDS (row stride 132 -> conflict-free
// ds_load_b64: lane bank index = 4*m + 2*h + k mod 64, all 32 lanes distinct pairs).
// Uniform control flow in the K loop: out-of-range rows are clamped (stores guarded),
// so EXEC stays all-1s into every V_WMMA (ISA requirement) with no saveexec churn.
__global__ void gemm128_wmma(const float* __restrict__ A, const float* __restrict__ B,
                             const float* __restrict__ bias, float* __restrict__ C,
                             long M, int do_relu) {
  __shared__ float As[16][132];
  int tid = threadIdx.x;
  long rowBase = (long)blockIdx.x * 16;

  // cooperative A-tile load: 16 rows x 128 cols, float4-coalesced (8 floats/thread)
  {
    int r = tid >> 4;            // 0..15
    int c = (tid & 15) * 8;      // 0..120 step 8
    long row = rowBase + r;
    if (row >= M) row = M - 1;   // clamp; garbage rows never stored
    const float4* srcp = (const float4*)(A + row * 128 + c);
    float4 v0 = srcp[0];
    float4 v1 = srcp[1];
    *(float4*)&As[r][c]     = v0;
    *(float4*)&As[r][c + 4] = v1;
  }
  __syncthreads();

  int wave = tid >> 5;
  int lane = tid & 31;
  int m = lane & 15;       // A row / B col / D col within tile
  int h = lane >> 4;       // selects K pair (A,B) and D row half
  int colBase = wave * 16;

  v8f acc = {};
#pragma unroll
  for (int k = 0; k < 128; k += 4) {
    int ka = k + 2 * h;
    v2f a = *(const v2f*)&As[m][ka];                      // A[row m][ka], [ka+1]
    v2f b;
    b.x = B[(long)ka * 128 + colBase + m];                // B[ka][col]
    b.y = B[(long)(ka + 1) * 128 + colBase + m];          // B[ka+1][col]
    acc = __builtin_amdgcn_wmma_f32_16x16x4_f32(
        /*neg_a=*/false, a, /*neg_b=*/false, b,
        /*c_mod=*/(short)0, acc, /*reuse_a=*/false, /*reuse_b=*/false);
  }
#pragma unroll
  for (int r = 0; r < 8; ++r) {
    long orow = rowBase + r + 8 * h;   // D: VGPR r, lane group h -> M = r + 8h
    int ocol = colBase + m;
    if (orow < M) {
      float v = acc[r];
      if (bias) v += bias[ocol];
      if (do_relu) v = fmaxf(v, 0.f);
      C[orow * 128 + ocol] = v;
    }
  }
}

// out[G,2] = g1[G,128] @ H2[128,2] + hb2
__global__ void head_out(const float* __restrict__ g1, const float* __restrict__ H2,
                         const float* __restrict__ hb2, float* __restrict__ out, long G) {
  long t = (long)blockIdx.x * blockDim.x + threadIdx.x;
  if (t >= G * 2) return;
  long i = t >> 1;
  int o = (int)(t & 1);
  float acc = hb2[o];
#pragma unroll 8
  for (int k = 0; k < 128; ++k) acc += g1[i * 128 + k] * H2[k * 2 + o];
  out[t] = acc;
}

// ---------------- host orchestration ----------------

static inline unsigned blocks_for(long n, int bs) { return (unsigned)((n + bs - 1) / bs); }

extern "C" void kernel_launch(void* const* d_in, const int* in_sizes, int n_in,
                              void* d_out, int out_size, void* d_ws, size_t ws_size,
                              hipStream_t stream) {
  const float* x    = (const float*)d_in[0];
  const int*   ei   = (const int*)d_in[1];
  const int*   batch= (const int*)d_in[2];
  const float* W1   = (const float*)d_in[4];
  const float* b1   = (const float*)d_in[5];
  const float* gn1w = (const float*)d_in[6];
  const float* gn1b = (const float*)d_in[7];
  const float* gn1a = (const float*)d_in[8];
  const float* W2   = (const float*)d_in[9];
  const float* b2   = (const float*)d_in[10];
  const float* gn2w = (const float*)d_in[11];
  const float* gn2b = (const float*)d_in[12];
  const float* gn2a = (const float*)d_in[13];
  const float* H1   = (const float*)d_in[14];
  const float* hb1  = (const float*)d_in[15];
  const float* H2   = (const float*)d_in[16];
  const float* hb2  = (const float*)d_in[17];
  float* out = (float*)d_out;

  long N = in_sizes[0] / 7;
  long E = in_sizes[1] / 2;
  long G = out_size / 2;

  const int* src = ei;
  const int* dst = ei + E;

  float* p = (float*)d_ws;
  auto alloc = [&](long n) { float* r = p; p += ((n + 63) / 64) * 64; return r; };
  float* bufA  = alloc(N * 128);
  float* bufB  = alloc(N * 128);
  float* dinv  = alloc(N);
  float* gsum  = alloc(G * 128);
  float* gmean = alloc(G * 128);
  float* gcnt  = alloc(G);
  float* pooled= alloc(G * 128);
  float* g1    = alloc(G * 128);
  (void)ws_size; (void)n_in;

  const int BS = 256;
  // degrees -> dinv, per-graph counts
  fill0<<<blocks_for(N, BS), BS, 0, stream>>>(dinv, N);
  deg_accum<<<blocks_for(E, BS), BS, 0, stream>>>(dst, dinv, E);
  dinv_k<<<blocks_for(N, BS), BS, 0, stream>>>(dinv, N);
  fill0<<<blocks_for(G, BS), BS, 0, stream>>>(gcnt, G);
  count_nodes<<<blocks_for(N, BS), BS, 0, stream>>>(batch, gcnt, N);

  // ---- Block 1: GCNConv(x, W1) -> GraphNorm -> ReLU ----
  gemm_in7<<<(unsigned)N, 128, 0, stream>>>(x, W1, bufA, N);
  fill0<<<blocks_for(N * 128, BS), BS, 0, stream>>>(bufB, N * 128);
  scatter_edges<<<blocks_for(E * 32, BS), BS, 0, stream>>>(src, dst, dinv, bufA, bufB, E);
  selfloop_bias<<<blocks_for(N * 128, BS), BS, 0, stream>>>(bufA, dinv, b1, bufB, N);

  fill0<<<blocks_for(G * 128, BS), BS, 0, stream>>>(gsum, G * 128);
  seg_accum<<<blocks_for(N * 32, BS), BS, 0, stream>>>(bufB, batch, gsum, N, 0);
  mean_div<<<blocks_for(G * 128, BS), BS, 0, stream>>>(gsum, gcnt, gmean, G);
  center_k<<<blocks_for(N * 128, BS), BS, 0, stream>>>(bufB, gmean, gn1a, batch, bufA, N);
  fill0<<<blocks_for(G * 128, BS), BS, 0, stream>>>(gsum, G * 128);
  seg_accum<<<blocks_for(N * 32, BS), BS, 0, stream>>>(bufA, batch, gsum, N, 1);
  gnorm_final<<<blocks_for(N * 128, BS), BS, 0, stream>>>(bufA, gsum, gcnt, gn1w, gn1b, batch, bufA, N);

  // ---- Block 2: GCNConv(y1, W2) via fp32 WMMA -> GraphNorm -> ReLU ----
  gemm128_wmma<<<blocks_for(N, 16), 256, 0, stream>>>(bufA, W2, nullptr, bufB, N, 0);
  fill0<<<blocks_for(N * 128, BS), BS, 0, stream>>>(bufA, N * 128);
  scatter_edges<<<blocks_for(E * 32, BS), BS, 0, stream>>>(src, dst, dinv, bufB, bufA, E);
  selfloop_bias<<<blocks_for(N * 128, BS), BS, 0, stream>>>(bufB, dinv, b2, bufA, N);

  fill0<<<blocks_for(G * 128, BS), BS, 0, stream>>>(gsum, G * 128);
  seg_accum<<<blocks_for(N * 32, BS), BS, 0, stream>>>(bufA, batch, gsum, N, 0);
  mean_div<<<blocks_for(G * 128, BS), BS, 0, stream>>>(gsum, gcnt, gmean, G);
  center_k<<<blocks_for(N * 128, BS), BS, 0, stream>>>(bufA, gmean, gn2a, batch, bufB, N);
  fill0<<<blocks_for(G * 128, BS), BS, 0, stream>>>(gsum, G * 128);
  seg_accum<<<blocks_for(N * 32, BS), BS, 0, stream>>>(bufB, batch, gsum, N, 1);
  gnorm_final<<<blocks_for(N * 128, BS), BS, 0, stream>>>(bufB, gsum, gcnt, gn2w, gn2b, batch, bufB, N);

  // ---- mean pool + MLP head ----
  fill0<<<blocks_for(G * 128, BS), BS, 0, stream>>>(pooled, G * 128);
  seg_accum<<<blocks_for(N * 32, BS), BS, 0, stream>>>(bufB, batch, pooled, N, 0);
  pool_div<<<blocks_for(G * 128, BS), BS, 0, stream>>>(pooled, gcnt, G);

  gemm128_wmma<<<blocks_for(G, 16), 256, 0, stream>>>(pooled, H1, hb1, g1, G, 1);
  head_out<<<blocks_for(G * 2, BS), BS, 0, stream>>>(g1, H2, hb2, out, G);
}